// OpenProtTransformerBlock_1520418423145
// MI455X (gfx1250) — compile-verified
//
#include <hip/hip_runtime.h>
#include <math.h>
#include <stdint.h>

// ---------------------------------------------------------------------------
// CDNA5 (gfx1250) wave32 fp32 WMMA implementation of the OpenProt block.
// Every GEMM lowers to V_WMMA_F32_16X16X4_F32 (one 16x16 C tile per wave).
// Triangle einsum uses async global->LDS staging (ASYNCcnt path).
// ---------------------------------------------------------------------------

typedef float v2f __attribute__((ext_vector_type(2)));
typedef float v8f __attribute__((ext_vector_type(8)));

#define LSEQ 512
#define DMODEL 512
#define PDIM 128
#define NH 8
#define DHD 64
#define FFD 2048
#define PFFD 512
#define LL 262144  // LSEQ*LSEQ
#define APAD 520   // padded LDS row stride (floats) to avoid bank conflicts

__device__ __forceinline__ v8f wmma4(v2f a, v2f b, v8f c) {
  // D = A(16x4,f32) * B(4x16,f32) + C(16x16,f32)
  return __builtin_amdgcn_wmma_f32_16x16x4_f32(false, a, false, b, (short)0, c,
                                               false, false);
}
__device__ __forceinline__ float sigm(float x) { return 1.f / (1.f + expf(-x)); }

__device__ __forceinline__ float block_reduce_sum(float v, float* red, int tid, int n) {
  red[tid] = v;
  __syncthreads();
  for (int s = n >> 1; s > 0; s >>= 1) {
    if (tid < s) red[tid] += red[tid + s];
    __syncthreads();
  }
  float r = red[0];
  __syncthreads();
  return r;
}

// ---------------------------------------------------------------------------
// LayerNorm over rows (generic), 256 threads per row.
// ---------------------------------------------------------------------------
__global__ void ln_kernel(const float* __restrict__ X, const float* __restrict__ g,
                          const float* __restrict__ b, float* __restrict__ Y, int ncols) {
  __shared__ float red[256];
  int row = blockIdx.x, tid = threadIdx.x;
  const float* xr = X + (size_t)row * ncols;
  float s = 0.f;
  for (int c = tid; c < ncols; c += 256) s += xr[c];
  float mean = block_reduce_sum(s, red, tid, 256) / (float)ncols;
  float v = 0.f;
  for (int c = tid; c < ncols; c += 256) { float d = xr[c] - mean; v += d * d; }
  float rinv = rsqrtf(block_reduce_sum(v, red, tid, 256) / (float)ncols + 1e-5f);
  float* yr = Y + (size_t)row * ncols;
  for (int c = tid; c < ncols; c += 256) yr[c] = (xr[c] - mean) * rinv * g[c] + b[c];
}

// ---------------------------------------------------------------------------
// Generic row-major WMMA GEMM: C = act(A*B + bias) (+ residual).
// ---------------------------------------------------------------------------
__global__ void gemm_kernel(const float* __restrict__ A, const float* __restrict__ B,
                            const float* __restrict__ bias,
                            const float* __restrict__ residual, float* __restrict__ C,
                            int M, int N, int K, int lda, int ldb, int ldc, int act) {
  int wave = threadIdx.x >> 5, lane = threadIdx.x & 31;
  int half = lane >> 4, lid = lane & 15;
  int tiles_n = N >> 4;
  int tid = blockIdx.x * 8 + wave;
  int total = (M >> 4) * tiles_n;
  if (tid >= total) return;
  int m0 = (tid / tiles_n) << 4, n0 = (tid % tiles_n) << 4;
  v8f acc = {0.f, 0.f, 0.f, 0.f, 0.f, 0.f, 0.f, 0.f};
  const float* ap = A + (size_t)(m0 + lid) * lda + 2 * half;
  const float* bp = B + (size_t)(2 * half) * ldb + n0 + lid;
  for (int k = 0; k < K; k += 4) {
    v2f a; a.x = ap[0]; a.y = ap[1];
    v2f b; b.x = bp[0]; b.y = bp[ldb];
    acc = wmma4(a, b, acc);
    ap += 4;
    bp += (size_t)4 * ldb;
  }
  for (int r = 0; r < 8; ++r) {
    int row = m0 + r + 8 * half, col = n0 + lid;
    float v = acc[r];
    if (bias) v += bias[col];
    if (act == 1) v = v > 0.f ? v : 0.f;
    size_t idx = (size_t)row * ldc + col;
    if (residual) v += residual[idx];
    C[idx] = v;
  }
}

// ---------------------------------------------------------------------------
// Pair bias: S[h,i,j] = LN(z[i,j,:]) . wzb[:,h] + bzb[h]
// ---------------------------------------------------------------------------
__global__ void pair_bias_kernel(const float* __restrict__ Z, const float* __restrict__ g,
                                 const float* __restrict__ b, const float* __restrict__ wzb,
                                 const float* __restrict__ bzb, float* __restrict__ S) {
  __shared__ float red[128];
  __shared__ float nv[128];
  int row = blockIdx.x, tid = threadIdx.x;
  float x = Z[(size_t)row * PDIM + tid];
  float mean = block_reduce_sum(x, red, tid, 128) * (1.f / 128.f);
  float d = x - mean;
  float rinv = rsqrtf(block_reduce_sum(d * d, red, tid, 128) * (1.f / 128.f) + 1e-5f);
  nv[tid] = d * rinv * g[tid] + b[tid];
  __syncthreads();
  for (int h = 0; h < NH; ++h) {
    float p = block_reduce_sum(nv[tid] * wzb[tid * NH + h], red, tid, 128);
    if (tid == 0) S[(size_t)h * LL + row] = p + bzb[h];
  }
}

// S[h,l,m] += scale * sum_d Q[l, h*64+d] * K[m, h*64+d]
__global__ void qk_kernel(const float* __restrict__ Q, const float* __restrict__ Km,
                          float* __restrict__ S, float scale) {
  int wave = threadIdx.x >> 5, lane = threadIdx.x & 31;
  int half = lane >> 4, lid = lane & 15;
  int h = blockIdx.y;
  int tid = blockIdx.x * 8 + wave;
  int m0 = (tid >> 5) << 4, n0 = (tid & 31) << 4;
  v8f acc = {0.f, 0.f, 0.f, 0.f, 0.f, 0.f, 0.f, 0.f};
  const float* ap = Q + (size_t)(m0 + lid) * DMODEL + h * DHD + 2 * half;
  const float* bp = Km + (size_t)(n0 + lid) * DMODEL + h * DHD + 2 * half;  // B^T
  for (int k = 0; k < DHD; k += 4) {
    v2f a; a.x = ap[0]; a.y = ap[1];
    v2f b; b.x = bp[0]; b.y = bp[1];
    acc = wmma4(a, b, acc);
    ap += 4; bp += 4;
  }
  for (int r = 0; r < 8; ++r) {
    int row = m0 + r + 8 * half, col = n0 + lid;
    size_t idx = (size_t)h * LL + (size_t)row * LSEQ + col;
    S[idx] += acc[r] * scale;
  }
}

__global__ void softmax_kernel(float* __restrict__ S, const int* __restrict__ mask) {
  __shared__ float red[256];
  int h = blockIdx.x >> 9, l = blockIdx.x & 511, tid = threadIdx.x;
  float* row = S + (size_t)h * LL + (size_t)l * LSEQ;
  float vals[2];
  float mx = -3.4e38f;
  for (int u = 0; u < 2; ++u) {
    int m = tid + u * 256;
    float v = (mask[m] != 0) ? row[m] : -1e9f;
    vals[u] = v;
    mx = fmaxf(mx, v);
  }
  red[tid] = mx; __syncthreads();
  for (int s = 128; s > 0; s >>= 1) {
    if (tid < s) red[tid] = fmaxf(red[tid], red[tid + s]);
    __syncthreads();
  }
  mx = red[0]; __syncthreads();
  float sum = 0.f;
  for (int u = 0; u < 2; ++u) { vals[u] = expf(vals[u] - mx); sum += vals[u]; }
  sum = block_reduce_sum(sum, red, tid, 256);
  float inv = 1.f / sum;
  for (int u = 0; u < 2; ++u) row[tid + u * 256] = vals[u] * inv;
}

// O[l, h*64+n] = sum_m A[h,l,m] * V[m, h*64+n]
__global__ void av_kernel(const float* __restrict__ S, const float* __restrict__ V,
                          float* __restrict__ O) {
  int wave = threadIdx.x >> 5, lane = threadIdx.x & 31;
  int half = lane >> 4, lid = lane & 15;
  int h = blockIdx.y;
  int tid = blockIdx.x * 8 + wave;
  int m0 = (tid >> 2) << 4, n0 = (tid & 3) << 4;
  v8f acc = {0.f, 0.f, 0.f, 0.f, 0.f, 0.f, 0.f, 0.f};
  const float* ap = S + (size_t)h * LL + (size_t)(m0 + lid) * LSEQ + 2 * half;
  const float* bp = V + (size_t)(2 * half) * DMODEL + h * DHD + n0 + lid;
  for (int k = 0; k < LSEQ; k += 4) {
    v2f a; a.x = ap[0]; a.y = ap[1];
    v2f b; b.x = bp[0]; b.y = bp[DMODEL];
    acc = wmma4(a, b, acc);
    ap += 4; bp += (size_t)4 * DMODEL;
  }
  for (int r = 0; r < 8; ++r) {
    int row = m0 + r + 8 * half, col = h * DHD + n0 + lid;
    O[(size_t)row * DMODEL + col] = acc[r];
  }
}

// ---------------------------------------------------------------------------
// seq2pair: z_out[i,j,:] = z_in[i,j,:] + [qh_i*kh_j | qh_i - kh_j] @ ow + ob
// ---------------------------------------------------------------------------
__global__ void s2p_kernel(const float* __restrict__ qk, const float* __restrict__ ow,
                           const float* __restrict__ ob, const float* __restrict__ zin,
                           float* __restrict__ zout) {
  __shared__ float As[16 * 128];
  int i = blockIdx.x >> 5;
  int j0 = (blockIdx.x & 31) << 4;
  int tid = threadIdx.x;
  for (int e = tid; e < 2048; e += 256) {
    int r = e >> 7, c = e & 127;
    int j = j0 + r;
    float v;
    if (c < 64) v = qk[i * PDIM + c] * qk[j * PDIM + 64 + c];
    else { int cc = c - 64; v = qk[i * PDIM + cc] - qk[j * PDIM + 64 + cc]; }
    As[e] = v;
  }
  __syncthreads();
  int wave = tid >> 5, lane = tid & 31, half = lane >> 4, lid = lane & 15;
  int n0 = wave << 4;
  v8f acc = {0.f, 0.f, 0.f, 0.f, 0.f, 0.f, 0.f, 0.f};
  for (int k = 0; k < 128; k += 4) {
    int kr = k + 2 * half;
    v2f a; a.x = As[lid * 128 + kr]; a.y = As[lid * 128 + kr + 1];
    v2f b; b.x = ow[kr * PDIM + n0 + lid]; b.y = ow[(kr + 1) * PDIM + n0 + lid];
    acc = wmma4(a, b, acc);
  }
  for (int r = 0; r < 8; ++r) {
    int j = j0 + r + 8 * half, col = n0 + lid;
    size_t idx = ((size_t)i * LSEQ + j) * PDIM + col;
    zout[idx] = zin[idx] + acc[r] + ob[col];
  }
}

// ---------------------------------------------------------------------------
// Triangle module: fused LN + 4 projections, writes a/b in (c,i,j) layout.
// ---------------------------------------------------------------------------
__global__ void tri_proj_kernel(const float* __restrict__ Z, const int* __restrict__ mask,
                                const float* __restrict__ lng, const float* __restrict__ lnb,
                                const float* __restrict__ agw, const float* __restrict__ agb,
                                const float* __restrict__ apw, const float* __restrict__ apb,
                                const float* __restrict__ bgw, const float* __restrict__ bgb,
                                const float* __restrict__ bpw, const float* __restrict__ bpb,
                                float* __restrict__ a_t, float* __restrict__ b_t) {
  __shared__ float As[16 * 128];
  __shared__ float mu[16], ri[16];
  int i = blockIdx.x >> 5;
  int j0 = (blockIdx.x & 31) << 4;
  int tid = threadIdx.x;
  for (int e = tid; e < 2048; e += 256) {
    int r = e >> 7, c = e & 127;
    As[e] = Z[((size_t)i * LSEQ + j0 + r) * PDIM + c];
  }
  __syncthreads();
  if (tid < 16) {
    float s = 0.f;
    for (int c = 0; c < 128; ++c) s += As[tid * 128 + c];
    float m = s * (1.f / 128.f);
    float v = 0.f;
    for (int c = 0; c < 128; ++c) { float d = As[tid * 128 + c] - m; v += d * d; }
    mu[tid] = m; ri[tid] = rsqrtf(v * (1.f / 128.f) + 1e-5f);
  }
  __syncthreads();
  for (int e = tid; e < 2048; e += 256) {
    int r = e >> 7, c = e & 127;
    As[e] = (As[e] - mu[r]) * ri[r] * lng[c] + lnb[c];
  }
  __syncthreads();
  int wave = tid >> 5, lane = tid & 31, half = lane >> 4, lid = lane & 15;
  int n0 = wave << 4;
  v8f aga = {0.f,0.f,0.f,0.f,0.f,0.f,0.f,0.f};
  v8f apa = aga, bga = aga, bpa = aga;
  for (int k = 0; k < 128; k += 4) {
    int kr = k + 2 * half;
    v2f a; a.x = As[lid * 128 + kr]; a.y = As[lid * 128 + kr + 1];
    v2f b;
    b.x = agw[kr * PDIM + n0 + lid]; b.y = agw[(kr + 1) * PDIM + n0 + lid];
    aga = wmma4(a, b, aga);
    b.x = apw[kr * PDIM + n0 + lid]; b.y = apw[(kr + 1) * PDIM + n0 + lid];
    apa = wmma4(a, b, apa);
    b.x = bgw[kr * PDIM + n0 + lid]; b.y = bgw[(kr + 1) * PDIM + n0 + lid];
    bga = wmma4(a, b, bga);
    b.x = bpw[kr * PDIM + n0 + lid]; b.y = bpw[(kr + 1) * PDIM + n0 + lid];
    bpa = wmma4(a, b, bpa);
  }
  float mfi = (float)mask[i];
  for (int r = 0; r < 8; ++r) {
    int j = j0 + r + 8 * half, c = n0 + lid;
    float tm = mfi * (float)mask[j];
    size_t idx = (size_t)c * LL + (size_t)i * LSEQ + j;
    a_t[idx] = sigm(aga[r] + agb[c]) * (apa[r] + apb[c]) * tm;
    b_t[idx] = sigm(bga[r] + bgb[c]) * (bpa[r] + bpb[c]) * tm;
  }
}

// ---------------------------------------------------------------------------
// Batched per-channel GEMM, block-cooperative:
//   outgoing: T_c = A_c * B_c^T ; incoming: T_c = A_c^T * B_c
// Each block owns one (channel, 16-row) slab: A staged ONCE into LDS with
// GLOBAL_LOAD_ASYNC_TO_LDS_B128 (ASYNCcnt), 8 waves x 4 accumulators cover
// the full 512-wide N range (A reuse x8 via LDS, x4 via accumulators).
// ---------------------------------------------------------------------------
__global__ void tri_gemm_kernel(const float* __restrict__ At, const float* __restrict__ Bt,
                                float* __restrict__ T, int outgoing) {
  __shared__ __align__(16) float As[16 * APAD];  // >= 512*16 too (k-major mode)
  int tid = threadIdx.x;
  int blk = blockIdx.x;
  int c = blk >> 5;          // 32 m-slabs per channel
  int m0 = (blk & 31) << 4;
  const float* Ac = At + (size_t)c * LL;
  const float* Bc = Bt + (size_t)c * LL;

  // ---- async stage of the A slab (32 KB) into LDS ----
  {
    const float* src = outgoing ? (Ac + (size_t)m0 * LSEQ) : (Ac + m0);
    unsigned lbase = (unsigned)(uintptr_t)(&As[0]);
    unsigned long long gbase = (unsigned long long)(uintptr_t)src;
    for (int e = tid; e < 2048; e += 256) {  // 2048 x 16B transfers
      unsigned lofs, gofs;
      if (outgoing) {            // row-major 16 x 512, padded rows in LDS
        int r = e >> 7, ch = e & 127;
        lofs = (unsigned)((r * APAD + ch * 4) * 4);
        gofs = (unsigned)((r * 512 + ch * 4) * 4);
      } else {                   // k-major 512 x 16 (A^T slab)
        int kk = e >> 2, ch = e & 3;
        lofs = (unsigned)((kk * 16 + ch * 4) * 4);
        gofs = (unsigned)((kk * 512 + ch * 4) * 4);
      }
      unsigned lds_addr = lbase + lofs;
      unsigned long long ga = gbase + gofs;
      asm volatile("global_load_async_to_lds_b128 %0, %1, off"
                   :: "v"(lds_addr), "v"(ga)
                   : "memory");
    }
    asm volatile("s_wait_asynccnt 0x0" ::: "memory");
  }
  __syncthreads();

  int wave = tid >> 5, lane = tid & 31, half = lane >> 4, lid = lane & 15;
  int nbase = wave << 6;  // 64 output columns per wave (4 tiles)
  v8f z8 = {0.f, 0.f, 0.f, 0.f, 0.f, 0.f, 0.f, 0.f};
  v8f acc0 = z8, acc1 = z8, acc2 = z8, acc3 = z8;

  if (outgoing) {
    const float* bp0 = Bc + (size_t)(nbase + lid) * LSEQ + 2 * half;  // B^T rows
    const float* bp1 = bp0 + (size_t)16 * LSEQ;
    const float* bp2 = bp0 + (size_t)32 * LSEQ;
    const float* bp3 = bp0 + (size_t)48 * LSEQ;
    for (int k = 0; k < LSEQ; k += 4) {
      int kr = k + 2 * half;
      v2f a; a.x = As[lid * APAD + kr]; a.y = As[lid * APAD + kr + 1];
      v2f b;
      b.x = bp0[0]; b.y = bp0[1]; acc0 = wmma4(a, b, acc0); bp0 += 4;
      b.x = bp1[0]; b.y = bp1[1]; acc1 = wmma4(a, b, acc1); bp1 += 4;
      b.x = bp2[0]; b.y = bp2[1]; acc2 = wmma4(a, b, acc2); bp2 += 4;
      b.x = bp3[0]; b.y = bp3[1]; acc3 = wmma4(a, b, acc3); bp3 += 4;
    }
  } else {
    const float* bp0 = Bc + (size_t)(2 * half) * LSEQ + nbase + lid;  // B row-major
    const float* bp1 = bp0 + 16;
    const float* bp2 = bp0 + 32;
    const float* bp3 = bp0 + 48;
    for (int k = 0; k < LSEQ; k += 4) {
      int kr = k + 2 * half;
      v2f a; a.x = As[kr * 16 + lid]; a.y = As[(kr + 1) * 16 + lid];
      v2f b;
      b.x = bp0[0]; b.y = bp0[LSEQ]; acc0 = wmma4(a, b, acc0); bp0 += 4 * LSEQ;
      b.x = bp1[0]; b.y = bp1[LSEQ]; acc1 = wmma4(a, b, acc1); bp1 += 4 * LSEQ;
      b.x = bp2[0]; b.y = bp2[LSEQ]; acc2 = wmma4(a, b, acc2); bp2 += 4 * LSEQ;
      b.x = bp3[0]; b.y = bp3[LSEQ]; acc3 = wmma4(a, b, acc3); bp3 += 4 * LSEQ;
    }
  }
  float* Tp = T + (size_t)c * LL;
  for (int r = 0; r < 8; ++r) {
    int row = m0 + r + 8 * half;
    float* o = Tp + (size_t)row * LSEQ + nbase + lid;
    o[0]  = acc0[r];
    o[16] = acc1[r];
    o[32] = acc2[r];
    o[48] = acc3[r];
  }
}

// (c,i,j) -> (i,j,c)
__global__ void transpose_kernel(const float* __restrict__ in, float* __restrict__ out) {
  size_t idx = (size_t)blockIdx.x * 256 + threadIdx.x;
  size_t c = idx >> 18;       // / LL
  size_t r = idx & (LL - 1);
  out[r * PDIM + c] = in[idx];
}

// z += sigmoid(LN_in(z)@g_w+g_b) * (LN_out(t)@z_w + z_b)
__global__ void tri_final_kernel(float* __restrict__ Zz, const float* __restrict__ Ttr,
                                 const float* __restrict__ lig, const float* __restrict__ lib,
                                 const float* __restrict__ log_, const float* __restrict__ lob,
                                 const float* __restrict__ gw, const float* __restrict__ gb,
                                 const float* __restrict__ zw, const float* __restrict__ zb) {
  __shared__ float ZL[16 * 128];
  __shared__ float TN[16 * 128];
  __shared__ float mu[32], ri[32];
  size_t row0 = (size_t)blockIdx.x * 16;
  int tid = threadIdx.x;
  for (int e = tid; e < 2048; e += 256) {
    int r = e >> 7, c = e & 127;
    ZL[e] = Zz[(row0 + r) * PDIM + c];
    TN[e] = Ttr[(row0 + r) * PDIM + c];
  }
  __syncthreads();
  if (tid < 32) {
    const float* src = (tid < 16) ? ZL : TN;
    int rr = tid & 15;
    float s = 0.f;
    for (int c = 0; c < 128; ++c) s += src[rr * 128 + c];
    float m = s * (1.f / 128.f);
    float v = 0.f;
    for (int c = 0; c < 128; ++c) { float d = src[rr * 128 + c] - m; v += d * d; }
    mu[tid] = m; ri[tid] = rsqrtf(v * (1.f / 128.f) + 1e-5f);
  }
  __syncthreads();
  for (int e = tid; e < 2048; e += 256) {
    int r = e >> 7, c = e & 127;
    ZL[e] = (ZL[e] - mu[r]) * ri[r] * lig[c] + lib[c];
    TN[e] = (TN[e] - mu[16 + r]) * ri[16 + r] * log_[c] + lob[c];
  }
  __syncthreads();
  int wave = tid >> 5, lane = tid & 31, half = lane >> 4, lid = lane & 15;
  int n0 = wave << 4;
  v8f ga = {0.f,0.f,0.f,0.f,0.f,0.f,0.f,0.f};
  v8f ta = ga;
  for (int k = 0; k < 128; k += 4) {
    int kr = k + 2 * half;
    v2f a; v2f b;
    a.x = ZL[lid * 128 + kr]; a.y = ZL[lid * 128 + kr + 1];
    b.x = gw[kr * PDIM + n0 + lid]; b.y = gw[(kr + 1) * PDIM + n0 + lid];
    ga = wmma4(a, b, ga);
    a.x = TN[lid * 128 + kr]; a.y = TN[lid * 128 + kr + 1];
    b.x = zw[kr * PDIM + n0 + lid]; b.y = zw[(kr + 1) * PDIM + n0 + lid];
    ta = wmma4(a, b, ta);
  }
  for (int r = 0; r < 8; ++r) {
    size_t row = row0 + r + 8 * half;
    int col = n0 + lid;
    size_t idx = row * PDIM + col;
    Zz[idx] = Zz[idx] + sigm(ga[r] + gb[col]) * (ta[r] + zb[col]);
  }
}

// ---------------------------------------------------------------------------
// Pair FFN, fully fused: z += relu(LN2(LN1(z))@w1+b1)@w2+b2, 16 rows/block.
// ---------------------------------------------------------------------------
__global__ void pair_ff_kernel(float* __restrict__ Zz,
                               const float* __restrict__ g1, const float* __restrict__ b1n,
                               const float* __restrict__ g2, const float* __restrict__ b2n,
                               const float* __restrict__ w1, const float* __restrict__ bb1,
                               const float* __restrict__ w2, const float* __restrict__ bb2) {
  __shared__ float As[16 * 128];
  __shared__ float Hs[16 * 512];
  __shared__ float mu[16], ri[16];
  size_t row0 = (size_t)blockIdx.x * 16;
  int tid = threadIdx.x;
  for (int e = tid; e < 2048; e += 256) {
    int r = e >> 7, c = e & 127;
    As[e] = Zz[(row0 + r) * PDIM + c];
  }
  __syncthreads();
  for (int pass = 0; pass < 2; ++pass) {
    if (tid < 16) {
      float s = 0.f;
      for (int c = 0; c < 128; ++c) s += As[tid * 128 + c];
      float m = s * (1.f / 128.f);
      float v = 0.f;
      for (int c = 0; c < 128; ++c) { float d = As[tid * 128 + c] - m; v += d * d; }
      mu[tid] = m; ri[tid] = rsqrtf(v * (1.f / 128.f) + 1e-5f);
    }
    __syncthreads();
    const float* gg = pass == 0 ? g1 : g2;
    const float* bb = pass == 0 ? b1n : b2n;
    for (int e = tid; e < 2048; e += 256) {
      int r = e >> 7, c = e & 127;
      As[e] = (As[e] - mu[r]) * ri[r] * gg[c] + bb[c];
    }
    __syncthreads();
  }
  int wave = tid >> 5, lane = tid & 31, half = lane >> 4, lid = lane & 15;
  for (int s = 0; s < 4; ++s) {
    int n0 = (wave * 4 + s) << 4;
    v8f acc = {0.f,0.f,0.f,0.f,0.f,0.f,0.f,0.f};
    for (int k = 0; k < 128; k += 4) {
      int kr = k + 2 * half;
      v2f a; a.x = As[lid * 128 + kr]; a.y = As[lid * 128 + kr + 1];
      v2f b; b.x = w1[kr * PFFD + n0 + lid]; b.y = w1[(kr + 1) * PFFD + n0 + lid];
      acc = wmma4(a, b, acc);
    }
    for (int r = 0; r < 8; ++r) {
      int row = r + 8 * half, col = n0 + lid;
      float v = acc[r] + bb1[col];
      Hs[row * 512 + col] = v > 0.f ? v : 0.f;
    }
  }
  __syncthreads();
  {
    int n0 = wave << 4;
    v8f acc = {0.f,0.f,0.f,0.f,0.f,0.f,0.f,0.f};
    for (int k = 0; k < 512; k += 4) {
      int kr = k + 2 * half;
      v2f a; a.x = Hs[lid * 512 + kr]; a.y = Hs[lid * 512 + kr + 1];
      v2f b; b.x = w2[kr * PDIM + n0 + lid]; b.y = w2[(kr + 1) * PDIM + n0 + lid];
      acc = wmma4(a, b, acc);
    }
    for (int r = 0; r < 8; ++r) {
      size_t row = row0 + r + 8 * half;
      int col = n0 + lid;
      size_t idx = row * PDIM + col;
      Zz[idx] = Zz[idx] + acc[r] + bb2[col];
    }
  }
}

// ---------------------------------------------------------------------------
// Frame update: fh = LN(x)@fu_w + fu_b; trans += R.vec; rots = rots @ R(rv)^T
// ---------------------------------------------------------------------------
__global__ void frame_kernel(const float* __restrict__ X, const float* __restrict__ g,
                             const float* __restrict__ b, const float* __restrict__ fw,
                             const float* __restrict__ fb, const float* __restrict__ rots,
                             const float* __restrict__ trans, float* __restrict__ rots_out,
                             float* __restrict__ trans_out) {
  __shared__ float red[64];
  __shared__ float sh[8];
  int l = blockIdx.x, tid = threadIdx.x;
  const float* xr = X + (size_t)l * DMODEL;
  float s = 0.f;
  for (int c = tid; c < DMODEL; c += 64) s += xr[c];
  float mean = block_reduce_sum(s, red, tid, 64) / (float)DMODEL;
  float v = 0.f;
  for (int c = tid; c < DMODEL; c += 64) { float d = xr[c] - mean; v += d * d; }
  float rinv = rsqrtf(block_reduce_sum(v, red, tid, 64) / (float)DMODEL + 1e-5f);
  float p[6] = {0.f, 0.f, 0.f, 0.f, 0.f, 0.f};
  for (int c = tid; c < DMODEL; c += 64) {
    float nv = (xr[c] - mean) * rinv * g[c] + b[c];
    for (int t = 0; t < 6; ++t) p[t] += nv * fw[c * 6 + t];
  }
  for (int t = 0; t < 6; ++t) {
    float r = block_reduce_sum(p[t], red, tid, 64);
    if (tid == 0) sh[t] = r + fb[t];
  }
  __syncthreads();
  if (tid == 0) {
    const float* R0 = rots + l * 9;
    float vx = sh[0], vy = sh[1], vz = sh[2];
    float rx = sh[3], ry = sh[4], rz = sh[5];
    for (int i = 0; i < 3; ++i)
      trans_out[l * 3 + i] =
          trans[l * 3 + i] + R0[i * 3 + 0] * vx + R0[i * 3 + 1] * vy + R0[i * 3 + 2] * vz;
    float ang = sqrtf(rx * rx + ry * ry + rz * rz);
    float dn = fmaxf(ang, 1e-8f);
    float ax = rx / dn, ay = ry / dn, az = rz / dn;
    float sA = sinf(ang), cA = cosf(ang), oc = 1.f - cA;
    float R[3][3];
    R[0][0] = cA + oc * ax * ax; R[0][1] = -sA * az + oc * ax * ay; R[0][2] = sA * ay + oc * ax * az;
    R[1][0] = sA * az + oc * ay * ax; R[1][1] = cA + oc * ay * ay; R[1][2] = -sA * ax + oc * ay * az;
    R[2][0] = -sA * ay + oc * az * ax; R[2][1] = sA * ax + oc * az * ay; R[2][2] = cA + oc * az * az;
    for (int r = 0; r < 3; ++r)
      for (int cc = 0; cc < 3; ++cc) {
        float a = 0.f;
        for (int k = 0; k < 3; ++k) a += R0[r * 3 + k] * R[cc][k];
        rots_out[l * 9 + r * 3 + cc] = a;
      }
  }
}

// ---------------------------------------------------------------------------
// Host side
// ---------------------------------------------------------------------------
enum {  // parameter order = dict insertion order in _make_params()
  WQ = 0, WK, WV, WO, WZB, FFI_W1, FFI_W2, S2P_W, S2P_OW, PFF_W1, PFF_W2, FU_W,
  TMO_AP_W, TMO_AG_W, TMO_BP_W, TMO_BG_W, TMO_G_W, TMO_Z_W,
  TMI_AP_W, TMI_AG_W, TMI_BP_W, TMI_BG_W, TMI_G_W, TMI_Z_W,
  BQ, BK, BV, BO, BZB, FFI_B1, FFI_B2, S2P_B, S2P_OB, PFF_B1, PFF_B2, FU_B,
  TMO_AP_B, TMO_AG_B, TMO_BP_B, TMO_BG_B, TMO_G_B, TMO_Z_B,
  TMI_AP_B, TMI_AG_B, TMI_BP_B, TMI_BG_B, TMI_G_B, TMI_Z_B,
  MHA_G, MHA_B, PAIRN_G, PAIRN_B, FFN_G, FFN_B, FFIN_G, FFIN_B,
  PFFN_G, PFFN_B, PAIRFFN_G, PAIRFFN_B, S2PLN_G, S2PLN_B, FULN_G, FULN_B,
  TMO_LNI_G, TMO_LNI_B, TMO_LNO_G, TMO_LNO_B, TMI_LNI_G, TMI_LNI_B, TMI_LNO_G, TMI_LNO_B
};

static void launch_tri(float* z, const int* mask, const float* const* P, int base_w,
                       int base_b, int base_ln, float* a_t, float* b_t, float* t_buf,
                       int outgoing, hipStream_t stream) {
  tri_proj_kernel<<<16384, 256, 0, stream>>>(
      z, mask, P[base_ln + 0], P[base_ln + 1],                       // ln_in g,b
      P[base_w + 1], P[base_b + 1],                                  // ag
      P[base_w + 0], P[base_b + 0],                                  // ap
      P[base_w + 3], P[base_b + 3],                                  // bg
      P[base_w + 2], P[base_b + 2],                                  // bp
      a_t, b_t);
  tri_gemm_kernel<<<4096, 256, 0, stream>>>(a_t, b_t, t_buf, outgoing);
  transpose_kernel<<<131072, 256, 0, stream>>>(t_buf, a_t);  // a_t now = t in (i,j,c)
  tri_final_kernel<<<16384, 256, 0, stream>>>(
      z, a_t, P[base_ln + 0], P[base_ln + 1], P[base_ln + 2], P[base_ln + 3],
      P[base_w + 4], P[base_b + 4], P[base_w + 5], P[base_b + 5]);
}

extern "C" void kernel_launch(void* const* d_in, const int* in_sizes, int n_in,
                              void* d_out, int out_size, void* d_ws, size_t ws_size,
                              hipStream_t stream) {
  const float* x_in = (const float*)d_in[0];
  const float* z_in = (const float*)d_in[1];
  const int* mask = (const int*)d_in[2];
  const float* rots_in = (const float*)d_in[3];
  const float* trans_in = (const float*)d_in[4];
  const float* const* P = (const float* const*)(d_in + 5);

  float* out = (float*)d_out;
  float* ox = out;                       // 512*512
  float* oz = out + (size_t)LSEQ * DMODEL;
  float* orot = oz + (size_t)LL * PDIM;
  float* otr = orot + (size_t)LSEQ * 9;

  float* ws = (float*)d_ws;
  float* w_h   = ws;                               // 262144
  float* w_buf = w_h + (size_t)LSEQ * DMODEL;      // 262144
  float* w_q   = w_buf + (size_t)LSEQ * DMODEL;
  float* w_k   = w_q + (size_t)LSEQ * DMODEL;
  float* w_v   = w_k + (size_t)LSEQ * DMODEL;
  float* w_o   = w_v + (size_t)LSEQ * DMODEL;
  float* w_ff  = w_o + (size_t)LSEQ * DMODEL;      // 512*2048
  float* w_s   = w_ff + (size_t)LSEQ * FFD;        // 8*512*512
  float* w_qk  = w_s + (size_t)NH * LL;            // 512*128
  float* w_A   = w_qk + (size_t)LSEQ * PDIM;       // LL*128  (a / t_transposed)
  float* w_B   = w_A + (size_t)LL * PDIM;          // LL*128  (b)
  float* w_T   = w_B + (size_t)LL * PDIM;          // LL*128  (t, channel-major)

  // ---- attention ----
  ln_kernel<<<LSEQ, 256, 0, stream>>>(x_in, P[MHA_G], P[MHA_B], w_h, DMODEL);
  pair_bias_kernel<<<LL, 128, 0, stream>>>(z_in, P[PAIRN_G], P[PAIRN_B], P[WZB], P[BZB], w_s);
  gemm_kernel<<<128, 256, 0, stream>>>(w_h, P[WQ], P[BQ], nullptr, w_q, 512, 512, 512, 512, 512, 512, 0);
  gemm_kernel<<<128, 256, 0, stream>>>(w_h, P[WK], P[BK], nullptr, w_k, 512, 512, 512, 512, 512, 512, 0);
  gemm_kernel<<<128, 256, 0, stream>>>(w_h, P[WV], P[BV], nullptr, w_v, 512, 512, 512, 512, 512, 512, 0);
  qk_kernel<<<dim3(128, NH), 256, 0, stream>>>(w_q, w_k, w_s, 0.125f);  // 1/sqrt(64)
  softmax_kernel<<<NH * LSEQ, 256, 0, stream>>>(w_s, mask);
  av_kernel<<<dim3(16, NH), 256, 0, stream>>>(w_s, w_v, w_o);
  gemm_kernel<<<128, 256, 0, stream>>>(w_o, P[WO], P[BO], x_in, ox, 512, 512, 512, 512, 512, 512, 0);

  // ---- x FFN (double LN) ----
  ln_kernel<<<LSEQ, 256, 0, stream>>>(ox, P[FFN_G], P[FFN_B], w_buf, DMODEL);
  ln_kernel<<<LSEQ, 256, 0, stream>>>(w_buf, P[FFIN_G], P[FFIN_B], w_h, DMODEL);
  gemm_kernel<<<512, 256, 0, stream>>>(w_h, P[FFI_W1], P[FFI_B1], nullptr, w_ff, 512, 2048, 512, 512, 2048, 2048, 1);
  gemm_kernel<<<128, 256, 0, stream>>>(w_ff, P[FFI_W2], P[FFI_B2], ox, ox, 512, 512, 2048, 2048, 512, 512, 0);

  // ---- seq -> pair ----
  ln_kernel<<<LSEQ, 256, 0, stream>>>(ox, P[S2PLN_G], P[S2PLN_B], w_buf, DMODEL);
  gemm_kernel<<<32, 256, 0, stream>>>(w_buf, P[S2P_W], P[S2P_B], nullptr, w_qk, 512, 128, 512, 512, 128, 128, 0);
  s2p_kernel<<<16384, 256, 0, stream>>>(w_qk, P[S2P_OW], P[S2P_OB], z_in, oz);

  // ---- triangle modules ----
  launch_tri(oz, mask, P, TMO_AP_W, TMO_AP_B, TMO_LNI_G, w_A, w_B, w_T, 1, stream);
  launch_tri(oz, mask, P, TMI_AP_W, TMI_AP_B, TMI_LNI_G, w_A, w_B, w_T, 0, stream);

  // ---- pair FFN (double LN, fused) ----
  pair_ff_kernel<<<16384, 256, 0, stream>>>(oz, P[PAIRFFN_G], P[PAIRFFN_B],
                                            P[PFFN_G], P[PFFN_B], P[PFF_W1], P[PFF_B1],
                                            P[PFF_W2], P[PFF_B2]);

  // ---- frame update ----
  frame_kernel<<<LSEQ, 64, 0, stream>>>(ox, P[FULN_G], P[FULN_B], P[FU_W], P[FU_B],
                                        rots_in, trans_in, orot, otr);

  (void)in_sizes; (void)n_in; (void)out_size; (void)ws_size;
}